// RotateModule_22789096472633
// MI455X (gfx1250) — compile-verified
//
#include <hip/hip_runtime.h>

// Y_a = R0^T * X_a * R1 for 16384 independent 64x64 fp32 matrices.
// Two chained 64x64x64 GEMMs per row via V_WMMA_F32_16X16X4_F32.
// R0/R1 fragments hoisted to registers (loop-invariant); X rows streamed
// HBM -> LDS with gfx1250 async loads (ASYNCcnt), double-buffered.

#define NDIM      64
#define ROW_ELEMS (NDIM * NDIM)   // 4096 floats per row

typedef float v2f __attribute__((ext_vector_type(2)));
typedef float v8f __attribute__((ext_vector_type(8)));

// Issue one row (16 KB) of async global->LDS b128 copies: 4 per thread.
__device__ __forceinline__ void async_copy_row(const float* __restrict__ gsrc,
                                               float* lds_dst, int tid)
{
#pragma unroll
    for (int i = 0; i < 4; ++i) {
        const float* g = gsrc + (size_t)(tid + i * 256) * 4;
        unsigned lds_off = (unsigned)(size_t)(lds_dst + (tid + i * 256) * 4);
        asm volatile("global_load_async_to_lds_b128 %0, %1, off"
                     :: "v"(lds_off), "v"(g)
                     : "memory");
    }
}

__device__ __forceinline__ void wait_async_zero()
{
    asm volatile("s_wait_asynccnt 0" ::: "memory");
}

__launch_bounds__(256, 1)
__global__ void rotate_wmma_kernel(const float* __restrict__ x,
                                   const float* __restrict__ R0,
                                   const float* __restrict__ R1,
                                   float* __restrict__ out,
                                   int rows)
{
    // Only X/T buffers live in LDS now: 2 x 16 KB.
    __shared__ __align__(16) float sX[2][ROW_ELEMS];

    const int tid  = threadIdx.x;
    const int lane = tid & 31;
    const int wave = tid >> 5;            // 0..7

    // WMMA fragment indexing (wave32):
    //  A 16x4 f32: lanes 0-15 -> K=0/1 in v0/v1, lanes 16-31 -> K=2/3. M = lane&15.
    //  B 4x16 f32: same K split, N = lane&15.
    //  C/D 16x16 f32: vgpr r -> M = r + (lane>=16 ? 8 : 0), N = lane&15.
    const int fm    = lane & 15;          // M (A) / N (B,C,D) within tile
    const int fk    = (lane >> 4) << 1;   // 0 or 2: base K of this lane-half
    const int rbase = (lane >> 4) << 3;   // 0 or 8: base M of C/D rows

    // ---- tile assignments (4x4 grid of 16x16 tiles, 2 per wave) ----
    // stage 1 (T = R0^T X): both tiles share jt  -> shared A (R0) frags
    const int s1_jt  = wave >> 1;
    const int s1_kt0 = (wave & 1) * 2;
    const int s1_kt1 = s1_kt0 + 1;
    // stage 2 (Y = T R1): both tiles share mt   -> shared B (R1) frags
    const int s2_mt  = wave >> 1;
    const int s2_jt0 = (wave & 1) * 2;
    const int s2_jt1 = s2_jt0 + 1;

    // ---- hoist loop-invariant R0 / R1 fragments into registers ----
    // A[m=j][K=i] = R0[i][j]  (stage-1 A frags for column block s1_jt)
    // B[K=k][n=m] = R1[k][m]  (stage-2 B frags for column block s2_mt)
    v2f aR0[16];
    v2f bR1[16];
#pragma unroll
    for (int i = 0; i < 16; ++i) {
        const int i0 = i * 4;
        aR0[i].x = R0[(i0 + fk    ) * NDIM + s1_jt * 16 + fm];
        aR0[i].y = R0[(i0 + fk + 1) * NDIM + s1_jt * 16 + fm];
        bR1[i].x = R1[(i0 + fk    ) * NDIM + s2_mt * 16 + fm];
        bR1[i].y = R1[(i0 + fk + 1) * NDIM + s2_mt * 16 + fm];
    }

    int row = blockIdx.x;
    int buf = 0;

    // Preload first row of X into buffer 0 (async HBM -> LDS).
    if (row < rows) {
        async_copy_row(x + (size_t)row * ROW_ELEMS, sX[0], tid);
    }
    wait_async_zero();
    __syncthreads();

    for (; row < rows; row += gridDim.x) {
        const int  nrow      = row + gridDim.x;
        const bool have_next = (nrow < rows);

        // Kick off async fill of the other buffer; overlaps with compute.
        if (have_next) {
            async_copy_row(x + (size_t)nrow * ROW_ELEMS, sX[buf ^ 1], tid);
        }

        const float* sXc = sX[buf];

        // ---------- stage 1: T = R0^T * X ----------
        v8f acc0 = {}, acc1 = {};
#pragma unroll
        for (int i = 0; i < 16; ++i) {
            const int i0 = i * 4;
            v2f b0, b1;
            b0.x = sXc[(i0 + fk    ) * NDIM + s1_kt0 * 16 + fm];
            b0.y = sXc[(i0 + fk + 1) * NDIM + s1_kt0 * 16 + fm];
            b1.x = sXc[(i0 + fk    ) * NDIM + s1_kt1 * 16 + fm];
            b1.y = sXc[(i0 + fk + 1) * NDIM + s1_kt1 * 16 + fm];
            acc0 = __builtin_amdgcn_wmma_f32_16x16x4_f32(false, aR0[i], false, b0,
                                                         (short)0, acc0, false, false);
            acc1 = __builtin_amdgcn_wmma_f32_16x16x4_f32(false, aR0[i], false, b1,
                                                         (short)0, acc1, false, false);
        }

        __syncthreads();   // all reads of X from sX[buf] complete

        // Write T tiles over the consumed X buffer.
        {
            float* sT = sX[buf];
#pragma unroll
            for (int r = 0; r < 8; ++r) {
                sT[(s1_jt * 16 + rbase + r) * NDIM + s1_kt0 * 16 + fm] = acc0[r];
                sT[(s1_jt * 16 + rbase + r) * NDIM + s1_kt1 * 16 + fm] = acc1[r];
            }
        }
        __syncthreads();   // T visible to all waves

        // ---------- stage 2: Y = T * R1 ----------
        {
            const float* sT = sX[buf];
            v8f c0 = {}, c1 = {};
#pragma unroll
            for (int i = 0; i < 16; ++i) {
                const int k0 = i * 4;
                // A frag rows are contiguous in T: aligned v2f loads.
                v2f a0 = *(const v2f*)(sT + (s2_jt0 * 16 + fm) * NDIM + k0 + fk);
                v2f a1 = *(const v2f*)(sT + (s2_jt1 * 16 + fm) * NDIM + k0 + fk);
                c0 = __builtin_amdgcn_wmma_f32_16x16x4_f32(false, a0, false, bR1[i],
                                                           (short)0, c0, false, false);
                c1 = __builtin_amdgcn_wmma_f32_16x16x4_f32(false, a1, false, bR1[i],
                                                           (short)0, c1, false, false);
            }

            float* orow = out + (size_t)row * ROW_ELEMS;
#pragma unroll
            for (int r = 0; r < 8; ++r) {
                orow[(s2_jt0 * 16 + rbase + r) * NDIM + s2_mt * 16 + fm] = c0[r];
                orow[(s2_jt1 * 16 + rbase + r) * NDIM + s2_mt * 16 + fm] = c1[r];
            }
        }

        // Next X buffer must be fully landed before the swap.
        wait_async_zero();
        __syncthreads();
        buf ^= 1;
    }
}

extern "C" void kernel_launch(void* const* d_in, const int* in_sizes, int n_in,
                              void* d_out, int out_size, void* d_ws, size_t ws_size,
                              hipStream_t stream)
{
    (void)n_in; (void)out_size; (void)d_ws; (void)ws_size;

    const float* x   = (const float*)d_in[0];
    const float* R0  = (const float*)d_in[1];
    const float* R1  = (const float*)d_in[2];
    float*       out = (float*)d_out;

    const int rows = in_sizes[0] / ROW_ELEMS;   // 16384
    int blocks = rows < 2048 ? rows : 2048;     // 8 rows per block at full size
    if (blocks < 1) blocks = 1;

    rotate_wmma_kernel<<<blocks, 256, 0, stream>>>(x, R0, R1, out, rows);
}